// MCAM_47407849013641
// MI455X (gfx1250) — compile-verified
//
#include <hip/hip_runtime.h>
#include <hip/hip_bf16.h>
#include <math.h>

// ---------------------------------------------------------------------------
// Shapes (fixed by the reference): B=8, C=512, H=W=64, HW=4096
// ---------------------------------------------------------------------------
#define BDIM 8
#define CDIM 512
#define WDIM 64
#define HWDIM 4096

typedef float v2f __attribute__((ext_vector_type(2)));
typedef float v8f __attribute__((ext_vector_type(8)));

__device__ __forceinline__ v8f wmma_f32(v2f a, v2f b, v8f c) {
    // V_WMMA_F32_16X16X4_F32 : D(16x16,f32) = A(16x4,f32) x B(4x16,f32) + C
    return __builtin_amdgcn_wmma_f32_16x16x4_f32(
        /*neg_a=*/false, a, /*neg_b=*/false, b,
        /*c_mod=*/(short)0, c, /*reuse_a=*/false, /*reuse_b=*/false);
}

// ---------------------------------------------------------------------------
// Async global->LDS (CDNA5: GLOBAL_LOAD_ASYNC_TO_LDS_B128, ASYNCcnt-tracked,
// bypasses VGPRs). The builtin's param types (from the compiler diagnostic):
//   (v4i __device__/AS1*, v4i AS3*, imm int offset, imm int cpol)
// Guarded: falls back to synchronous float4 staging if not declared.
// ---------------------------------------------------------------------------
#if __has_builtin(__builtin_amdgcn_global_load_async_to_lds_b128)
#define HAVE_ASYNC_LDS 1
#else
#define HAVE_ASYNC_LDS 0
#endif

typedef int v4i __attribute__((vector_size(16)));
typedef __attribute__((address_space(1))) v4i* gas_v4i_ptr;   // global AS
typedef __attribute__((address_space(3))) v4i* lds_v4i_ptr;   // LDS AS

__device__ __forceinline__ void copy16_g2l(const float* g, float* l) {
#if HAVE_ASYNC_LDS
    v4i* gv = (v4i*)g;   // retype (drops const) in generic AS
    v4i* lv = (v4i*)l;
    __builtin_amdgcn_global_load_async_to_lds_b128(
        (gas_v4i_ptr)gv, (lds_v4i_ptr)lv, /*imm offset=*/0, /*cpol=*/0);
#else
    *(float4*)l = *(const float4*)g;
#endif
}

__device__ __forceinline__ void wait_async_lds() {
#if HAVE_ASYNC_LDS
#if __has_builtin(__builtin_amdgcn_s_wait_asynccnt)
    __builtin_amdgcn_s_wait_asynccnt(0);
#else
    asm volatile("s_wait_asynccnt 0" ::: "memory");
#endif
#endif
}

// ---------------------------------------------------------------------------
// Kernel 1: 1x1 conv as GEMM.  F[b] = Wc(512x512) * X[b](512x4096)
// Block: 128 threads (4 waves). Each block computes a 64(M) x 64(N) tile.
// Wave wv owns N-subtile wv and all 4 M-subtiles -> 4 v8f accumulators.
// K blocked by 32, double-buffered through LDS with async global->LDS copies:
// tile k+1 streams in while WMMAs consume tile k.
// LDS pad strides: A stride 36 floats, B stride 80 floats -> fragment reads
// are bank-conflict-free (lane halves hit disjoint bank groups) and rows are
// 16B-multiples (144B / 320B) so B128 async writes stay aligned.
// ---------------------------------------------------------------------------
__global__ __launch_bounds__(128) void k_conv1x1(const float* __restrict__ Wc,
                                                 const float* __restrict__ X,
                                                 float* __restrict__ F) {
    __shared__ float a_lds[2][64][36];   // W tile: 64 rows(M) x 32 cols(K)
    __shared__ float b_lds[2][32][80];   // X tile: 32 rows(K) x 64 cols(N)

    const int t    = threadIdx.x;
    const int lane = t & 31;
    const int wv   = t >> 5;
    const int n0   = blockIdx.x * 64;        // pixel tile
    const int m0   = blockIdx.y * 64;        // out-channel tile
    const int b    = blockIdx.z;             // batch

    const float* Xb = X + (size_t)b * CDIM * HWDIM;
    float*       Fb = F + (size_t)b * CDIM * HWDIM;

    const int lrow = lane & 15;              // M (A) / N (B) index
    const int khi  = (lane >> 4) * 2;        // K base for this lane half

    v8f acc[4];
    #pragma unroll
    for (int mt = 0; mt < 4; ++mt)
        #pragma unroll
        for (int e = 0; e < 8; ++e) acc[mt][e] = 0.0f;

    // 8 x 16B transfers per thread per K-block (4 for A tile, 4 for B tile)
    auto load_tile = [&](int buf, int kb) {
        #pragma unroll
        for (int i = t; i < 512; i += 128) {
            const int m  = i >> 3;
            const int k4 = (i & 7) * 4;
            copy16_g2l(Wc + (size_t)(m0 + m) * CDIM + kb + k4, &a_lds[buf][m][k4]);
        }
        #pragma unroll
        for (int i = t; i < 512; i += 128) {
            const int k  = i >> 4;
            const int n4 = (i & 15) * 4;
            copy16_g2l(Xb + (size_t)(kb + k) * HWDIM + n0 + n4, &b_lds[buf][k][n4]);
        }
    };

    load_tile(0, 0);

    for (int it = 0; it < CDIM / 32; ++it) {
        const int cur = it & 1;
        wait_async_lds();       // current buffer's async copies complete (this wave)
        __syncthreads();        // ... and for every wave in the group

        if (it + 1 < CDIM / 32)
            load_tile(cur ^ 1, (it + 1) * 32);   // prefetch overlaps the WMMAs below

        #pragma unroll
        for (int kk = 0; kk < 32; kk += 4) {
            v2f bf;
            bf.x = b_lds[cur][kk + khi + 0][wv * 16 + lrow];
            bf.y = b_lds[cur][kk + khi + 1][wv * 16 + lrow];
            #pragma unroll
            for (int mt = 0; mt < 4; ++mt) {
                v2f af;
                af.x = a_lds[cur][mt * 16 + lrow][kk + khi + 0];
                af.y = a_lds[cur][mt * 16 + lrow][kk + khi + 1];
                acc[mt] = wmma_f32(af, bf, acc[mt]);
            }
        }
        __syncthreads();        // everyone done reading `cur` before it is refilled
    }

    // C/D layout: VGPR v -> M = v + 8*(lane>>4), N = lane&15
    const int col   = n0 + wv * 16 + lrow;
    const int rbase = m0 + 8 * (lane >> 4);
    #pragma unroll
    for (int mt = 0; mt < 4; ++mt)
        #pragma unroll
        for (int v = 0; v < 8; ++v)
            Fb[(size_t)(rbase + mt * 16 + v) * HWDIM + col] = acc[mt][v];
}

// ---------------------------------------------------------------------------
// Kernel 2: per-(b,c) Gram matrix  S = F^T F  with F the 64x64 [h][w] plane.
// One block (4 waves) per (b,c). Plane staged in LDS via async copies
// (stride 80 floats: bank-conflict-free fragment reads, 320B rows for B128).
// D = A^T A means A-frag and B-frag share the same load pattern
// (elem = F[k][16*tile + lane%16]); wave wv owns row-tile I=wv, J=0..3.
// ---------------------------------------------------------------------------
__global__ __launch_bounds__(128) void k_gram(const float* __restrict__ Fin,
                                              float* __restrict__ Sout) {
    __shared__ float f_lds[64][80];

    const int t    = threadIdx.x;
    const int lane = t & 31;
    const int wv   = t >> 5;

    const float* P = Fin + (size_t)blockIdx.x * HWDIM;  // blockIdx.x in [0, B*C)
    float*       Q = Sout + (size_t)blockIdx.x * HWDIM;

    #pragma unroll
    for (int i = t; i < 1024; i += 128) {
        const int h  = i >> 4;
        const int w4 = (i & 15) * 4;
        copy16_g2l(P + h * WDIM + w4, &f_lds[h][w4]);
    }
    wait_async_lds();
    __syncthreads();

    const int lrow = lane & 15;
    const int khi  = (lane >> 4) * 2;

    v8f acc[4];
    #pragma unroll
    for (int j = 0; j < 4; ++j)
        #pragma unroll
        for (int e = 0; e < 8; ++e) acc[j][e] = 0.0f;

    #pragma unroll
    for (int kk = 0; kk < 64; kk += 4) {
        v2f af;
        af.x = f_lds[kk + khi + 0][wv * 16 + lrow];
        af.y = f_lds[kk + khi + 1][wv * 16 + lrow];
        #pragma unroll
        for (int J = 0; J < 4; ++J) {
            v2f bf;
            bf.x = f_lds[kk + khi + 0][J * 16 + lrow];
            bf.y = f_lds[kk + khi + 1][J * 16 + lrow];
            acc[J] = wmma_f32(af, bf, acc[J]);
        }
    }

    const int rb = wv * 16 + 8 * (lane >> 4);
    #pragma unroll
    for (int J = 0; J < 4; ++J)
        #pragma unroll
        for (int v = 0; v < 8; ++v)
            Q[(rb + v) * WDIM + J * 16 + lrow] = acc[J][v];
}

// ---------------------------------------------------------------------------
// Kernel 3: softmax over channel axis (dim=1) for both branches + fused
// combine:  Att = V_o * V_s * (softmax(S_o) * softmax(S_s))^2
// Block = (b, i) row; 256 threads = 64 j-positions x 4 channel groups.
// Online softmax per (b,i,j); partials merged through LDS; second pass
// re-reads S (L2-resident: S_o+S_s = 128MB < 192MB L2) and writes the output.
// All accesses are 256B-coalesced rows.
// ---------------------------------------------------------------------------
__global__ __launch_bounds__(256) void k_softmax_combine(
    const float* __restrict__ Sopt, const float* __restrict__ Ssar,
    const float* __restrict__ Vopt, const float* __restrict__ Vsar,
    float* __restrict__ Out) {
    __shared__ float sm_o[4][64], sl_o[4][64], sm_s[4][64], sl_s[4][64];

    const int i  = blockIdx.x;   // 0..63
    const int b  = blockIdx.y;   // 0..7
    const int t  = threadIdx.x;
    const int j  = t & 63;
    const int cg = t >> 6;       // 0..3

    const size_t base = ((size_t)b * CDIM) * HWDIM + (size_t)i * WDIM + j;

    float mo = -INFINITY, lo = 0.0f, ms = -INFINITY, ls = 0.0f;
    for (int c = cg; c < CDIM; c += 4) {
        const size_t off = base + (size_t)c * HWDIM;
        float vo = Sopt[off];
        float m2 = fmaxf(mo, vo);
        lo = lo * __expf(mo - m2) + __expf(vo - m2);
        mo = m2;
        float vs = Ssar[off];
        m2 = fmaxf(ms, vs);
        ls = ls * __expf(ms - m2) + __expf(vs - m2);
        ms = m2;
    }
    sm_o[cg][j] = mo; sl_o[cg][j] = lo;
    sm_s[cg][j] = ms; sl_s[cg][j] = ls;
    __syncthreads();

    float Mo = fmaxf(fmaxf(sm_o[0][j], sm_o[1][j]), fmaxf(sm_o[2][j], sm_o[3][j]));
    float Ms = fmaxf(fmaxf(sm_s[0][j], sm_s[1][j]), fmaxf(sm_s[2][j], sm_s[3][j]));
    float Lo = 0.0f, Ls = 0.0f;
    #pragma unroll
    for (int g = 0; g < 4; ++g) {
        Lo += sl_o[g][j] * __expf(sm_o[g][j] - Mo);
        Ls += sl_s[g][j] * __expf(sm_s[g][j] - Ms);
    }
    const float rLo = 1.0f / Lo;
    const float rLs = 1.0f / Ls;

    for (int c = cg; c < CDIM; c += 4) {
        const size_t off = base + (size_t)c * HWDIM;
        const float po  = __expf(Sopt[off] - Mo) * rLo;
        const float ps  = __expf(Ssar[off] - Ms) * rLs;
        const float had = po * ps;
        Out[off] = Vopt[off] * Vsar[off] * had * had;
    }
}

// ---------------------------------------------------------------------------
// Launch: conv1x1 (x2 branches) -> gram (x2) -> fused softmax/combine.
// Workspace: f_opt | f_sar | s_opt | s_sar  (4 x 64MB fp32 = 256MB)
// ---------------------------------------------------------------------------
extern "C" void kernel_launch(void* const* d_in, const int* in_sizes, int n_in,
                              void* d_out, int out_size, void* d_ws, size_t ws_size,
                              hipStream_t stream) {
    const float* opt  = (const float*)d_in[0];
    const float* sar  = (const float*)d_in[1];
    const float* Wopt = (const float*)d_in[2];
    const float* Wsar = (const float*)d_in[3];
    float* out = (float*)d_out;

    const size_t N = (size_t)BDIM * CDIM * HWDIM;  // 16,777,216
    float* f_opt = (float*)d_ws;
    float* f_sar = f_opt + N;
    float* s_opt = f_sar + N;
    float* s_sar = s_opt + N;

    dim3 g1(HWDIM / 64, CDIM / 64, BDIM);  // (64, 8, 8)
    k_conv1x1<<<g1, 128, 0, stream>>>(Wopt, opt, f_opt);
    k_conv1x1<<<g1, 128, 0, stream>>>(Wsar, sar, f_sar);

    k_gram<<<BDIM * CDIM, 128, 0, stream>>>(f_opt, s_opt);
    k_gram<<<BDIM * CDIM, 128, 0, stream>>>(f_sar, s_sar);

    dim3 g3(WDIM, BDIM);  // (64, 8)
    k_softmax_combine<<<g3, 256, 0, stream>>>(s_opt, s_sar, f_opt, f_sar, out);
}